// DWT_Features_9972914061877
// MI455X (gfx1250) — compile-verified
//
#include <hip/hip_runtime.h>
#include <hip/hip_bf16.h>

// ---------------------------------------------------------------------------
// DWT_Features: x[8192,1,64,8,8] --(3-level db4 DWT over t + conv3d + bias +
// leakyrelu)--> out[8192,3].  Everything before the LeakyReLU is linear, so we
// fold the DWT matrix M[84,64] into conv_w to get Wpad[16][4096] (K padded
// 3->16) and run one memory-bound GEMM with V_WMMA_F32_16X16X4_F32.
// Split-K x4 + 4 rotating accumulators break the WMMA RAW chain so the kernel
// stays on the 23.3 TB/s HBM roofline instead of the XDL latency chain.
// ---------------------------------------------------------------------------

typedef __attribute__((ext_vector_type(2))) float v2f;
typedef __attribute__((ext_vector_type(8))) float v8f;

// db4 analysis filters, reversed (cross-correlation form used by the reference)
__device__ __constant__ float HLO_REV[8] = {
     0.23037781330885523f,  0.7148465705525415f,   0.6308807679295904f,
    -0.02798376941698385f, -0.18703481171888114f,  0.030841381835986965f,
     0.032883011666982945f, -0.010597401784997278f };
__device__ __constant__ float HHI_REV[8] = {
    -0.010597401784997278f, 0.032883011666982945f, 0.030841381835986965f,
     0.18703481171888114f, -0.02798376941698385f, -0.6308807679295904f,
     0.7148465705525415f,  -0.23037781330885523f };

__device__ __forceinline__ int refl(int q, int n) {
    if (q < 0)  q = -q;            // reflect (no edge repeat), single bounce
    if (q >= n) q = 2 * n - 2 - q; // is enough: pads <= 7 < n-1 for n >= 14
    return q;
}

__device__ void afb_level(const float* x, int n, float* lo, float* hi) {
    const int out = (n + 7) >> 1;
    const int p   = 2 * (out - 1) - n + 8;
    const int pl  = p >> 1;
    for (int m = 0; m < out; ++m) {
        float aL = 0.f, aH = 0.f;
        for (int j = 0; j < 8; ++j) {
            float v = x[refl(2 * m + j - pl, n)];
            aL += v * HLO_REV[j];
            aH += v * HHI_REV[j];
        }
        lo[m] = aL;
        hi[m] = aH;
    }
}

// One block, 256 threads. Threads 0..63 each push a unit vector e_t through the
// 3-level DWT to build column t of M[84][64] in LDS; then all threads fold M
// with conv_w[3,84,8,8] into wpad[16][4096] (rows 3..15 zero).
__global__ void __launch_bounds__(256)
dwt_build_weights(const float* __restrict__ conv_w, float* __restrict__ wpad) {
    __shared__ float M[84 * 64];
    const int tid = threadIdx.x;

    if (tid < 64) {
        float sig[64];
        for (int i = 0; i < 64; ++i) sig[i] = (i == tid) ? 1.f : 0.f;
        float lo1[35], hi1[35];
        afb_level(sig, 64, lo1, hi1);
        float lo2[21], hi2[21];
        afb_level(lo1, 35, lo2, hi2);
        float lo3[14], hi3[14];
        afb_level(lo2, 21, lo3, hi3);
        // coeff order: [lo3(14), hi1(35), hi2(21), hi3(14)]
        for (int c = 0; c < 14; ++c) M[(c      ) * 64 + tid] = lo3[c];
        for (int c = 0; c < 35; ++c) M[(14 + c ) * 64 + tid] = hi1[c];
        for (int c = 0; c < 21; ++c) M[(49 + c ) * 64 + tid] = hi2[c];
        for (int c = 0; c < 14; ++c) M[(70 + c ) * 64 + tid] = hi3[c];
    }
    __syncthreads();

    // wpad[k][t*64 + s] = sum_c M[c][t] * conv_w[k, c, s],  s = h*8+w
    for (int idx = tid; idx < 16 * 4096; idx += 256) {
        const int k    = idx >> 12;
        const int r    = idx & 4095;
        const int t    = r >> 6;
        const int sidx = r & 63;
        float acc = 0.f;
        if (k < 3) {
            const float* wk = conv_w + k * (84 * 64) + sidx;
            for (int c = 0; c < 84; ++c)
                acc += M[c * 64 + t] * wk[c * 64];
        }
        wpad[idx] = acc;
    }
}

// 256 threads = 8 waves per block: 2 batch-tiles x 4 K-slices.
// Each wave: D(16x16) += A(16x4) x B(4x16) over its 1024-column K slice,
// using 4 rotating accumulators; partials reduced through LDS, then
// bias + LeakyReLU on the D layout (lane&15 = channel, only n<3 stores).
__global__ void __launch_bounds__(256)
dwt_fused_gemm(const float* __restrict__ x, const float* __restrict__ wpad,
               const float* __restrict__ bias, float* __restrict__ out) {
    __shared__ float red[8][256];             // [wave][v*32 + lane]

    const int lane  = threadIdx.x & 31;
    const int wave  = threadIdx.x >> 5;
    const int tloc  = wave >> 2;              // tile within block: 0 or 1
    const int kslc  = wave & 3;               // K slice: 0..3
    const int tile  = blockIdx.x * 2 + tloc;  // 0..511
    const int b0    = tile * 16;
    const int m     = lane & 15;              // row-in-tile (A) / channel (B,D)
    const int hf    = lane >> 4;              // lane half -> K pair {0,1}/{2,3}
    const int coff  = 2 * hf;
    const int kbase = kslc * 1024;

    const float* xp = x    + (size_t)(b0 + m) * 4096 + kbase + coff;
    const float* wp = wpad + (size_t)m        * 4096 + kbase + coff;

    v8f a0 = {}, a1 = {}, a2 = {}, a3 = {};
#pragma unroll 4
    for (int i = 0; i < 1024; i += 16) {
        v2f xa0 = *(const v2f*)(xp + i);
        v2f wb0 = *(const v2f*)(wp + i);
        v2f xa1 = *(const v2f*)(xp + i + 4);
        v2f wb1 = *(const v2f*)(wp + i + 4);
        v2f xa2 = *(const v2f*)(xp + i + 8);
        v2f wb2 = *(const v2f*)(wp + i + 8);
        v2f xa3 = *(const v2f*)(xp + i + 12);
        v2f wb3 = *(const v2f*)(wp + i + 12);
        a0 = __builtin_amdgcn_wmma_f32_16x16x4_f32(false, xa0, false, wb0,
                                                   (short)0, a0, false, false);
        a1 = __builtin_amdgcn_wmma_f32_16x16x4_f32(false, xa1, false, wb1,
                                                   (short)0, a1, false, false);
        a2 = __builtin_amdgcn_wmma_f32_16x16x4_f32(false, xa2, false, wb2,
                                                   (short)0, a2, false, false);
        a3 = __builtin_amdgcn_wmma_f32_16x16x4_f32(false, xa3, false, wb3,
                                                   (short)0, a3, false, false);
    }
    v8f acc = (a0 + a1) + (a2 + a3);

    // Stash this wave's partial D, then reduce the 4 K-slices per tile.
#pragma unroll
    for (int v = 0; v < 8; ++v)
        red[wave][v * 32 + lane] = acc[v];
    __syncthreads();

    if (kslc == 0) {                          // waves 0 and 4 finalize a tile
        const float bv = bias[m < 3 ? m : 0];
        if (m < 3) {
#pragma unroll
            for (int v = 0; v < 8; ++v) {
                float f = red[wave][v * 32 + lane] +
                          red[wave + 1][v * 32 + lane] +
                          red[wave + 2][v * 32 + lane] +
                          red[wave + 3][v * 32 + lane] + bv;
                const int brow = b0 + v + 8 * hf;   // D: VGPR v, half -> +8
                out[brow * 3 + m] = (f >= 0.f) ? f : 0.001f * f;
            }
        }
    }
}

extern "C" void kernel_launch(void* const* d_in, const int* in_sizes, int n_in,
                              void* d_out, int out_size, void* d_ws, size_t ws_size,
                              hipStream_t stream) {
    const float* x      = (const float*)d_in[0];   // [8192,1,64,8,8]
    const float* conv_w = (const float*)d_in[1];   // [3,84,8,8]
    const float* conv_b = (const float*)d_in[2];   // [3]
    float*       out    = (float*)d_out;           // [8192,3]
    float*       wpad   = (float*)d_ws;            // 16*4096 floats = 256 KB

    hipLaunchKernelGGL(dwt_build_weights, dim3(1), dim3(256), 0, stream,
                       conv_w, wpad);
    hipLaunchKernelGGL(dwt_fused_gemm, dim3(256), dim3(256), 0, stream,
                       x, wpad, conv_b, out);
}